// MyEdgeConv_61194694033729
// MI455X (gfx1250) — compile-verified
//
#include <hip/hip_runtime.h>

#define B_    8
#define C_    3
#define N_    4096
#define K_    20
#define H_    64
#define DOUT_ 64
#define EPS_  1e-5f

typedef __attribute__((ext_vector_type(2))) float v2f;
typedef __attribute__((ext_vector_type(8))) float v8f;

// -------------------------------------------------------------------------
// Kernel 1: brute-force KNN (negative squared distance, top-K=20).
// One thread per query point; the whole batch's point cloud (48 KB) is
// staged in LDS (320 KB/WGP on CDNA5 makes this cheap).
// -------------------------------------------------------------------------
__global__ __launch_bounds__(256) void knn_kernel(const float* __restrict__ x,
                                                  int* __restrict__ idx)
{
    __shared__ float sx[N_], sy[N_], sz[N_];
    const int blocksPerBatch = N_ / 256;
    const int b     = blockIdx.x / blocksPerBatch;
    const int chunk = blockIdx.x % blocksPerBatch;
    const float* xb = x + (size_t)b * 3 * N_;
    for (int j = threadIdx.x; j < N_; j += 256) {
        sx[j] = xb[j];
        sy[j] = xb[N_ + j];
        sz[j] = xb[2 * N_ + j];
    }
    __syncthreads();

    const int   n  = chunk * 256 + threadIdx.x;
    const float qx = sx[n], qy = sy[n], qz = sz[n];
    const float qq = qx * qx + qy * qy + qz * qz;

    float bd[K_];
    int   bi[K_];
#pragma unroll
    for (int t = 0; t < K_; ++t) { bd[t] = -3.4e38f; bi[t] = 0; }

    for (int j = 0; j < N_; ++j) {
        const float px = sx[j], py = sy[j], pz = sz[j];
        const float d = 2.0f * (qx * px + qy * py + qz * pz) - qq
                        - (px * px + py * py + pz * pz);
        if (d > bd[K_ - 1]) {             // strict > keeps lower index on ties
            bd[K_ - 1] = d; bi[K_ - 1] = j;
#pragma unroll
            for (int t = K_ - 1; t > 0; --t) {
                if (bd[t] > bd[t - 1]) {
                    float tf = bd[t]; bd[t] = bd[t - 1]; bd[t - 1] = tf;
                    int   ti = bi[t]; bi[t] = bi[t - 1]; bi[t - 1] = ti;
                }
            }
        }
    }
    int* outp = idx + ((size_t)b * N_ + n) * K_;
#pragma unroll
    for (int t = 0; t < K_; ++t) outp[t] = bi[t];
}

// -------------------------------------------------------------------------
// Kernel 2: edge conv (gf[32pad,8pad] x w1^T[8pad,64]) via f32 WMMA,
// fused BN + ReLU + max/mean pooling over the 20 neighbors.
// One wave (32 lanes) per point. 16 v_wmma_f32_16x16x4_f32 per wave.
// All loads are unconditional (indices clamped, values masked by selects)
// so no EXEC-mask branches appear in the WMMA-dense region.
// -------------------------------------------------------------------------
__global__ __launch_bounds__(128) void edge_kernel(
    const float* __restrict__ x,  const int* __restrict__ idx,
    const float* __restrict__ w1, const float* __restrict__ b1,
    const float* __restrict__ g1, const float* __restrict__ beta1,
    const float* __restrict__ rm1,const float* __restrict__ rv1,
    float* __restrict__ out_m1,   float* __restrict__ out_m2)
{
    const int lane = threadIdx.x & 31;
    const int half = lane >> 4;   // 0: K-cols {0,1}/{4,5}; 1: {2,3}/{6,7}
    const int l15  = lane & 15;
    const int wave = (blockIdx.x * blockDim.x + threadIdx.x) >> 5;
    const int b    = wave >> 12;          // / N_
    const int n    = wave & (N_ - 1);

    const float* xb = x + (size_t)b * 3 * N_;
    const float cx = xb[n], cy = xb[N_ + n], cz = xb[2 * N_ + n];

    // A tiles: row-tile rt covers neighbors rt*16 .. rt*16+15 (>=20 padded 0)
    v2f a0[2], a1[2];
#pragma unroll
    for (int rt = 0; rt < 2; ++rt) {
        const int  kk    = rt * 16 + l15;        // row M = l15 in this tile
        const bool valid = (kk < K_);
        const int  kc    = valid ? kk : (K_ - 1);     // clamp -> uncond. load
        const int  nb    = idx[((size_t)b * N_ + n) * K_ + kc];
        const float nx = xb[nb], ny = xb[N_ + nb], nz = xb[2 * N_ + nb];
        const float e0 = valid ? (nx - cx) : 0.0f;   // gf col 0
        const float e1 = valid ? (ny - cy) : 0.0f;   // gf col 1
        const float e2 = valid ? (nz - cz) : 0.0f;   // gf col 2
        const float e3 = valid ? cx : 0.0f;          // gf col 3
        const float e4 = valid ? cy : 0.0f;          // gf col 4
        const float e5 = valid ? cz : 0.0f;          // gf col 5
        v2f t0, t1;
        t0.x = half ? e2 : e0;  t0.y = half ? e3 : e1;
        t1.x = half ? 0.0f : e4; t1.y = half ? 0.0f : e5;  // cols 6,7 = 0
        a0[rt] = t0; a1[rt] = t1;
    }

    // 4 column tiles of 16 output channels; 2 k-steps each
    v8f acc0[4], acc1[4];
#pragma unroll
    for (int ct = 0; ct < 4; ++ct) {
        const int    o    = ct * 16 + l15;
        const float* wrow = w1 + o * 6;
        // unconditional loads, scalar selects (pure v_cndmask, no exec mask)
        const float w0 = wrow[0], w1v = wrow[1], w2v = wrow[2];
        const float w3 = wrow[3], w4v = wrow[4], w5v = wrow[5];
        v2f bt0, bt1;
        bt0.x = half ? w2v : w0;   bt0.y = half ? w3 : w1v;
        bt1.x = half ? 0.0f : w4v; bt1.y = half ? 0.0f : w5v;
        v8f z = {};
        z = __builtin_amdgcn_wmma_f32_16x16x4_f32(false, a0[0], false, bt0,
                                                  (short)0, z, false, false);
        z = __builtin_amdgcn_wmma_f32_16x16x4_f32(false, a1[0], false, bt1,
                                                  (short)0, z, false, false);
        acc0[ct] = z;
        v8f z1 = {};
        z1 = __builtin_amdgcn_wmma_f32_16x16x4_f32(false, a0[1], false, bt0,
                                                   (short)0, z1, false, false);
        z1 = __builtin_amdgcn_wmma_f32_16x16x4_f32(false, a1[1], false, bt1,
                                                   (short)0, z1, false, false);
        acc1[ct] = z1;
    }

    // BN + ReLU + pooling. Column o lives in lanes l15 and l15+16;
    // D rows: acc0 -> rows v+8*half (all valid), acc1 -> rows 16+v+8*half
    // (valid only when half==0 && v<4, i.e. neighbors 16..19).
#pragma unroll
    for (int ct = 0; ct < 4; ++ct) {
        const int   o = ct * 16 + l15;
        const float s = g1[o] * rsqrtf(rv1[o] + EPS_);
        const float t = (b1[o] - rm1[o]) * s + beta1[o];
        float mx = -3.4e38f, sm = 0.0f;
#pragma unroll
        for (int v = 0; v < 8; ++v) {
            const float val = fmaxf(acc0[ct][v] * s + t, 0.0f);
            mx = fmaxf(mx, val); sm += val;
        }
#pragma unroll
        for (int v = 0; v < 4; ++v) {
            const float val = fmaxf(acc1[ct][v] * s + t, 0.0f);
            mx = (half == 0) ? fmaxf(mx, val) : mx;
            sm = (half == 0) ? (sm + val) : sm;
        }
        mx = fmaxf(mx, __shfl_xor(mx, 16, 32));
        sm = sm + __shfl_xor(sm, 16, 32);
        if (half == 0) {
            const size_t off = ((size_t)b * 64 + o) * N_ + n;
            out_m1[off] = mx;
            out_m2[off] = sm * (1.0f / K_);
        }
    }
}

// -------------------------------------------------------------------------
// Kernel 3: [B*N,128] x w2^T[128,64] via f32 WMMA + BN + ReLU.
// One wave per 16-row tile; fully unrolled: 32 k-steps x 4 col-tiles =
// 128 v_wmma_f32_16x16x4_f32 per wave, w2 staged in 32 KB LDS.
// A is read from the m1/m2 output sections ([B,64,N] layout).
// -------------------------------------------------------------------------
__global__ __launch_bounds__(128) void conv2_kernel(
    const float* __restrict__ m1, const float* __restrict__ m2,
    const float* __restrict__ w2, const float* __restrict__ b2,
    const float* __restrict__ g2, const float* __restrict__ beta2,
    const float* __restrict__ rm2,const float* __restrict__ rv2,
    float* __restrict__ out)
{
    __shared__ float sw[DOUT_ * 128];          // 32 KB
    for (int i = threadIdx.x; i < DOUT_ * 128; i += 128) sw[i] = w2[i];
    __syncthreads();

    const int lane    = threadIdx.x & 31;
    const int half    = lane >> 4;
    const int l15     = lane & 15;
    const int wave    = (blockIdx.x * blockDim.x + threadIdx.x) >> 5;
    const int rowbase = wave * 16;

    const int m  = rowbase + l15;              // A row owned by this lane
    const int bb = m >> 12;
    const int nn = m & (N_ - 1);
    const float* arow1 = m1 + (size_t)bb * 64 * N_ + nn;   // + c*N_
    const float* arow2 = m2 + (size_t)bb * 64 * N_ + nn;

    v8f acc[4] = {{}, {}, {}, {}};
#pragma unroll
    for (int ks = 0; ks < 32; ++ks) {          // ks uniform across the wave
        const int k0 = ks * 4 + half * 2;
        float av0, av1;
        if (ks < 16) {                          // compile-time after unroll
            av0 = arow1[(size_t)k0 * N_];
            av1 = arow1[(size_t)(k0 + 1) * N_];
        } else {
            av0 = arow2[(size_t)(k0 - 64) * N_];
            av1 = arow2[(size_t)(k0 - 63) * N_];
        }
        const v2f a = {av0, av1};
#pragma unroll
        for (int ct = 0; ct < 4; ++ct) {
            const int o = ct * 16 + l15;
            const v2f bt = { sw[o * 128 + k0], sw[o * 128 + k0 + 1] };
            acc[ct] = __builtin_amdgcn_wmma_f32_16x16x4_f32(
                false, a, false, bt, (short)0, acc[ct], false, false);
        }
    }

#pragma unroll
    for (int ct = 0; ct < 4; ++ct) {
        const int   o = ct * 16 + l15;
        const float s = g2[o] * rsqrtf(rv2[o] + EPS_);
        const float t = (b2[o] - rm2[o]) * s + beta2[o];
#pragma unroll
        for (int v = 0; v < 8; ++v) {
            const int mr = rowbase + v + half * 8;
            const int ob = mr >> 12;
            const int on = mr & (N_ - 1);
            out[((size_t)ob * 64 + o) * N_ + on] = fmaxf(acc[ct][v] * s + t, 0.0f);
        }
    }
}

// -------------------------------------------------------------------------
extern "C" void kernel_launch(void* const* d_in, const int* in_sizes, int n_in,
                              void* d_out, int out_size, void* d_ws, size_t ws_size,
                              hipStream_t stream)
{
    const float* x     = (const float*)d_in[0];
    const float* w1    = (const float*)d_in[1];
    const float* b1    = (const float*)d_in[2];
    const float* g1    = (const float*)d_in[3];
    const float* beta1 = (const float*)d_in[4];
    const float* rm1   = (const float*)d_in[5];
    const float* rv1   = (const float*)d_in[6];
    const float* w2    = (const float*)d_in[7];
    const float* b2    = (const float*)d_in[8];
    const float* g2    = (const float*)d_in[9];
    const float* beta2 = (const float*)d_in[10];
    const float* rm2   = (const float*)d_in[11];
    const float* rv2   = (const float*)d_in[12];

    float* out    = (float*)d_out;
    float* out_m1 = out + (size_t)B_ * 64 * N_;       // tuple element 1
    float* out_m2 = out + 2 * (size_t)B_ * 64 * N_;   // tuple element 2
    int*   idx    = (int*)d_ws;                       // B*N*K ints = 2.62 MB

    knn_kernel<<<B_ * (N_ / 256), 256, 0, stream>>>(x, idx);

    // one wave per point, 4 waves per block
    edge_kernel<<<(B_ * N_) / 4, 128, 0, stream>>>(
        x, idx, w1, b1, g1, beta1, rm1, rv1, out_m1, out_m2);

    // one wave per 16-row tile: (B*N/16) waves, 4 per block
    conv2_kernel<<<(B_ * N_ / 16) / 4, 128, 0, stream>>>(
        out_m1, out_m2, w2, b2, g2, beta2, rm2, rv2, out);
}